// ProNetMasifLigand_37108517438327
// MI455X (gfx1250) — compile-verified
//
#include <hip/hip_runtime.h>

typedef __attribute__((ext_vector_type(2))) float v2f;
typedef __attribute__((ext_vector_type(8))) float v8f;

#define B_  16
#define V_  8192
#define N_  2048
#define H_  128
#define CO_ 7
#define MT_ 4   // vertex tiles (of 16) per wave in the argmin kernel

// ---------------------------------------------------------------------------
// Kernel 0: zero hit mask + pooled accumulator, precompute residue norms ||c||^2
// ---------------------------------------------------------------------------
__global__ void pnml_init(const float* __restrict__ coords,
                          float* __restrict__ hit,
                          float* __restrict__ cn2,
                          float* __restrict__ pooled) {
  int tid = blockIdx.x * blockDim.x + threadIdx.x;
  if (tid < B_ * N_) {
    hit[tid] = 0.0f;
    float x = coords[tid * 3 + 0];
    float y = coords[tid * 3 + 1];
    float z = coords[tid * 3 + 2];
    cn2[tid] = x * x + y * y + z * z;
  }
  if (tid < B_ * H_) pooled[tid] = 0.0f;
}

// ---------------------------------------------------------------------------
// Kernel 1: WMMA-tiled nearest-residue argmin.
// One wave handles 64 vertices (4 A tiles); loops over 128 residue tiles,
// issuing 4 independent V_WMMA_F32_16X16X4_F32 per B tile.
// Score folded entirely into the WMMA:  A' = -2*v,  C = ||c||^2 broadcast,
// so D = ||c||^2 - 2 v.c  (the argmin score) comes straight out of the
// matrix pipe with zero trailing FMAs. All loads are branch-free.
// ---------------------------------------------------------------------------
__global__ void pnml_argmin(const float* __restrict__ verts,
                            const float* __restrict__ coords,
                            const float* __restrict__ cn2,
                            float* __restrict__ hit) {
  const int wave = (blockIdx.x * blockDim.x + threadIdx.x) >> 5;
  const int lane = threadIdx.x & 31;
  const int half = lane >> 4;      // 0: lanes 0-15 (K=0,1), 1: lanes 16-31 (K=2,3)
  const int lm   = lane & 15;

  const int b    = wave >> 7;      // 128 waves (of 64 verts) per batch
  const int tile = wave & 127;
  const int m0   = tile * (16 * MT_);

  const float* __restrict__ vb  = verts  + (size_t)b * V_ * 3;
  const float* __restrict__ cb  = coords + (size_t)b * N_ * 3;
  const float* __restrict__ c2b = cn2    + (size_t)b * N_;

  // A tiles: 16x4 f32 layout, pre-scaled by -2. lanes 0-15 hold (K0,K1)=
  // (-2x,-2y); lanes 16-31 hold (K2,K3)=(-2z,0). Branch-free loads.
  v2f a[MT_];
#pragma unroll
  for (int t = 0; t < MT_; ++t) {
    const int m = m0 + t * 16 + lm;
    const float ax = vb[m * 3 + 2 * half];      // lower: x, upper: z
    const float ay = vb[m * 3 + 1 + half];      // lower: y, upper: z(dropped)
    a[t].x = -2.0f * ax;
    a[t].y = half ? 0.0f : (-2.0f * ay);
  }

  float bestd[MT_][8];
  int   bestn[MT_][8];
#pragma unroll
  for (int t = 0; t < MT_; ++t)
#pragma unroll
    for (int r = 0; r < 8; ++r) { bestd[t][r] = 3.0e38f; bestn[t][r] = 0; }

  for (int j = 0; j < N_ / 16; ++j) {
    const int n = j * 16 + lm;
    // B tile 4x16 f32 (mirror layout), branch-free loads
    const float bx = cb[n * 3 + 2 * half];      // lower: x, upper: z
    const float by = cb[n * 3 + 1 + half];      // lower: y, upper: z(dropped)
    v2f bv;
    bv.x = bx;
    bv.y = half ? 0.0f : by;
    const float c2 = c2b[n];

    // Broadcast ||c||^2 into the accumulator: D = -2 v.c + ||c||^2 = score
    v8f cc;
#pragma unroll
    for (int r = 0; r < 8; ++r) cc[r] = c2;

    v8f d[MT_];
#pragma unroll
    for (int t = 0; t < MT_; ++t) {
      d[t] = __builtin_amdgcn_wmma_f32_16x16x4_f32(
          false, a[t], false, bv, (short)0, cc, false, false);
    }

    // D layout: lane column = lm; VGPR r -> M = r + 8*half
#pragma unroll
    for (int t = 0; t < MT_; ++t) {
#pragma unroll
      for (int r = 0; r < 8; ++r) {
        const float score = d[t][r];
        if (score < bestd[t][r]) { bestd[t][r] = score; bestn[t][r] = n; }
      }
    }
  }

  // Reduce min/argmin across the 16 lanes of each half (masks < 16 stay in-half)
#pragma unroll
  for (int t = 0; t < MT_; ++t) {
#pragma unroll
    for (int r = 0; r < 8; ++r) {
      float bd = bestd[t][r];
      int   bn = bestn[t][r];
#pragma unroll
      for (int off = 8; off >= 1; off >>= 1) {
        const float od = __shfl_xor(bd, off, 32);
        const int   on = __shfl_xor(bn, off, 32);
        if (od < bd) { bd = od; bn = on; }
      }
      if (lm == 0) hit[b * N_ + bn] = 1.0f;   // benign same-value races
    }
  }
}

// ---------------------------------------------------------------------------
// Kernel 2: masked pooling  pooled[b,h] += sum_n hit[b,n] * feats[b,n,h]
// Grid: (B * 16) blocks, each covers a 128-residue chunk; 128 threads = H lanes
// ---------------------------------------------------------------------------
__global__ void pnml_pool(const float* __restrict__ feats,
                          const float* __restrict__ hit,
                          float* __restrict__ pooled) {
  const int b     = blockIdx.x >> 4;
  const int chunk = blockIdx.x & 15;
  const int h     = threadIdx.x;           // 0..127
  const int n0    = chunk * 128;

  __shared__ float msk[128];
  msk[h] = hit[b * N_ + n0 + h];
  __syncthreads();

  const float* __restrict__ fb = feats + ((size_t)b * N_ + n0) * H_ + h;
  float s = 0.0f;
  for (int i = 0; i < 128; ++i) s += msk[i] * fb[(size_t)i * H_];
  atomicAdd(&pooled[b * H_ + h], s);
}

// ---------------------------------------------------------------------------
// Kernel 3: MLP head. pooled[16,128] @ W1[128,128] via WMMA (M=16 = batch dim
// is exactly one tile; 8 waves x 16 columns, 32 K-steps of 4), +b1, ReLU ->
// LDS, then tiny second GEMM [16,128]@[128,7]+b2 scalar.
// ---------------------------------------------------------------------------
__global__ void pnml_mlp(const float* __restrict__ pooled,
                         const float* __restrict__ W1,
                         const float* __restrict__ b1,
                         const float* __restrict__ W2,
                         const float* __restrict__ b2,
                         float* __restrict__ out) {
  __shared__ float h1[16][H_ + 1];

  const int lane = threadIdx.x & 31;
  const int w    = threadIdx.x >> 5;       // 8 waves
  const int half = lane >> 4;
  const int lm   = lane & 15;
  const int col  = w * 16 + lm;
  const int k0   = 2 * half;               // lower half: K0,K1; upper: K2,K3

  v8f acc = {};
  for (int kt = 0; kt < H_ / 4; ++kt) {
    const int k4 = kt * 4 + k0;
    v2f a, bv;
    a.x  = pooled[lm * H_ + k4 + 0];
    a.y  = pooled[lm * H_ + k4 + 1];
    bv.x = W1[(k4 + 0) * H_ + col];
    bv.y = W1[(k4 + 1) * H_ + col];
    acc = __builtin_amdgcn_wmma_f32_16x16x4_f32(
        false, a, false, bv, (short)0, acc, false, false);
  }

  const float bias = b1[col];
#pragma unroll
  for (int r = 0; r < 8; ++r) {
    const int m = r + half * 8;
    const float v = acc[r] + bias;
    h1[m][col] = v > 0.0f ? v : 0.0f;
  }
  __syncthreads();

  const int t = threadIdx.x;
  if (t < 16 * CO_) {
    const int m = t / CO_;
    const int c = t % CO_;
    float s = b2[c];
    for (int j = 0; j < H_; ++j) s += h1[m][j] * W2[j * CO_ + c];
    out[m * CO_ + c] = s;
  }
}

// ---------------------------------------------------------------------------
extern "C" void kernel_launch(void* const* d_in, const int* in_sizes, int n_in,
                              void* d_out, int out_size, void* d_ws, size_t ws_size,
                              hipStream_t stream) {
  const float* verts  = (const float*)d_in[0];   // [B,V,3]
  const float* coords = (const float*)d_in[1];   // [B,N,3]
  const float* feats  = (const float*)d_in[2];   // [B,N,H]
  const float* W1     = (const float*)d_in[3];   // [H,H]
  const float* b1     = (const float*)d_in[4];   // [H]
  const float* W2     = (const float*)d_in[5];   // [H,CO]
  const float* b2     = (const float*)d_in[6];   // [CO]
  float* out = (float*)d_out;                    // [B,CO]

  float* ws     = (float*)d_ws;
  float* hit    = ws;                        // B*N floats
  float* cn2    = ws + (size_t)B_ * N_;      // B*N floats
  float* pooled = ws + (size_t)2 * B_ * N_;  // B*H floats

  // 0) init: zero hit/pooled, precompute ||c||^2
  pnml_init<<<(B_ * N_ + 255) / 256, 256, 0, stream>>>(coords, hit, cn2, pooled);

  // 1) WMMA argmin: 2048 waves (64 vertices each), 8 waves/block
  pnml_argmin<<<(B_ * (V_ / (16 * MT_))) / 8, 256, 0, stream>>>(verts, coords, cn2, hit);

  // 2) masked pooling
  pnml_pool<<<B_ * 16, 128, 0, stream>>>(feats, hit, pooled);

  // 3) MLP head (single block, 8 waves)
  pnml_mlp<<<1, 256, 0, stream>>>(pooled, W1, b1, W2, b2, out);
}